// Glm4moeMoe_47356309405776
// MI455X (gfx1250) — compile-verified
//
#include <hip/hip_runtime.h>
#include <hip/hip_bf16.h>

// Problem dims (match reference)
#define TT 512
#define HH 2048
#define EE 64
#define KTOP 8
#define II 768
#define CHUNK 128   // H-chunk staged in LDS for stage-1 GEMMs

typedef float v2f __attribute__((ext_vector_type(2)));
typedef float v8f __attribute__((ext_vector_type(8)));
typedef unsigned int v4u __attribute__((ext_vector_type(4)));
typedef int v4i __attribute__((ext_vector_type(4)));
typedef int v8i __attribute__((ext_vector_type(8)));

#define V8F_ZERO {0.f,0.f,0.f,0.f,0.f,0.f,0.f,0.f}

#if __has_builtin(__builtin_amdgcn_tensor_load_to_lds) && \
    __has_builtin(__builtin_amdgcn_s_wait_tensorcnt)
#define USE_TDM 1
#else
#define USE_TDM 0
#endif

static __device__ __forceinline__ v8f wmma4(v2f a, v2f b, v8f c) {
  // D = A(16x4,f32) * B(4x16,f32) + C(16x16,f32)
  return __builtin_amdgcn_wmma_f32_16x16x4_f32(
      /*neg_a=*/false, a, /*neg_b=*/false, b,
      /*c_mod=*/(short)0, c, /*reuse_a=*/false, /*reuse_b=*/false);
}

static __device__ __forceinline__ float fast_rcp(float v) {
#if __has_builtin(__builtin_amdgcn_rcpf)
  return __builtin_amdgcn_rcpf(v);
#else
  return 1.0f / v;
#endif
}

static __device__ __forceinline__ float sigmoid_f(float v) {
  return fast_rcp(1.0f + __expf(-v));
}

#if USE_TDM
// ---------------------------------------------------------------------------
// Tensor Data Mover helpers (CDNA5 ISA ch.8: D# groups).
// Group1 packer for a 2D tile of 4-byte elements, optional LDS row padding.
// ---------------------------------------------------------------------------
static __device__ __forceinline__ v8i tdm_group1_2d(
    unsigned tensor_d0, unsigned tensor_d1, unsigned tile_d0, unsigned tile_d1,
    unsigned long long stride0, unsigned pad_int_code, unsigned pad_amt_code,
    bool pad_en) {
  v8i g;
  unsigned w0 = (2u << 16);                  // data_size = 2 -> 4 bytes
  if (pad_en)
    w0 |= (1u << 20) | (pad_int_code << 22) | (pad_amt_code << 25);
  g[0] = (int)w0;                            // wg_mask=0, no atomic barrier
  g[1] = (int)((tensor_d0 & 0xFFFFu) << 16); // [63:48] = tensor_dim0 lo16
  g[2] = (int)((tensor_d0 >> 16) | ((tensor_d1 & 0xFFFFu) << 16));
  g[3] = (int)((tensor_d1 >> 16) | (tile_d0 << 16));     // [127:112]=tile_dim0
  g[4] = (int)(tile_d1 & 0xFFFFu);                        // tile_dim2 = 0
  g[5] = (int)(unsigned)(stride0 & 0xFFFFFFFFull);        // dim0_stride lo32
  g[6] = (int)(unsigned)((stride0 >> 32) & 0xFFFFull);    // dim0_stride hi16
  g[7] = 0;                                               // dim1_stride = 0
  return g;
}

static __device__ __forceinline__ void tdm_issue(
    unsigned lds_off, const void* gaddr, unsigned flags,
    v8i g1, v4i g2, v4i g3) {
  unsigned long long ga = (unsigned long long)(size_t)gaddr;
  v4u g0;
  g0[0] = 1u | flags;                        // count=1 (+gather mode bits)
  g0[1] = lds_off;                           // LDS byte address
  g0[2] = (unsigned)(ga & 0xFFFFFFFFu);      // global_addr[31:0]
  g0[3] = (unsigned)((ga >> 32) & 0x01FFFFFFu) | (2u << 30);  // type=2
#if defined(__clang_major__) && (__clang_major__ >= 23)
  v8i g4 = {0, 0, 0, 0, 0, 0, 0, 0};
  __builtin_amdgcn_tensor_load_to_lds(g0, g1, g2, g3, g4, 0);
#else
  __builtin_amdgcn_tensor_load_to_lds(g0, g1, g2, g3, 0);
#endif
}
#endif  // USE_TDM

// ---------------------------------------------------------------------------
// Router phase 1: scores[t,e] = sigmoid(x[t] . gate_w[e])
// ---------------------------------------------------------------------------
__global__ __launch_bounds__(64)
void router_scores_kernel(const float* __restrict__ x,
                          const float* __restrict__ gate_w,
                          float* __restrict__ scores) {
  const int t = blockIdx.x;
  const int e = threadIdx.x;
  const float4* xr = (const float4*)(x + (size_t)t * HH);
  const float4* wr = (const float4*)(gate_w + (size_t)e * HH);
  float acc = 0.0f;
  for (int i = 0; i < HH / 4; ++i) {
    float4 a = xr[i];
    float4 b = wr[i];
    acc += a.x * b.x + a.y * b.y + a.z * b.z + a.w * b.w;
  }
  scores[(size_t)t * EE + e] = sigmoid_f(acc);
}

__global__ void zero_counts_kernel(int* __restrict__ cnt) {
  if (threadIdx.x < EE) cnt[threadIdx.x] = 0;
}

// ---------------------------------------------------------------------------
// Router phase 2: per-token top-8 on (scores + bias); per-expert list build.
// ---------------------------------------------------------------------------
__global__ __launch_bounds__(256)
void router_topk_kernel(const float* __restrict__ scores,
                        const float* __restrict__ gate_b,
                        int* __restrict__ expert_count,
                        int* __restrict__ tok_list,
                        float* __restrict__ wt_list) {
  const int t = blockIdx.x * blockDim.x + threadIdx.x;
  if (t >= TT) return;
  const float* srow = scores + (size_t)t * EE;

  unsigned long long used = 0ull;
  int   sel_e[KTOP];
  float sel_w[KTOP];
  float wsum = 0.0f;

  for (int k = 0; k < KTOP; ++k) {
    float best = -1e30f;
    int   be   = 0;
    for (int e = 0; e < EE; ++e) {
      if (used & (1ull << e)) continue;
      float b = srow[e] + gate_b[e];
      if (b > best) { best = b; be = e; }
    }
    used |= (1ull << be);
    sel_e[k] = be;
    float raw = srow[be];
    sel_w[k] = raw;
    wsum += raw;
  }
  const float inv = 1.0f / (wsum + 1e-20f);  // ROUTED_SCALING == 1.0

  for (int k = 0; k < KTOP; ++k) {
    int e   = sel_e[k];
    int pos = atomicAdd(&expert_count[e], 1);
    tok_list[(size_t)e * TT + pos] = t;
    wt_list[(size_t)e * TT + pos] = sel_w[k] * inv;
  }
}

// ---------------------------------------------------------------------------
// Shared expert: out[t0..t0+16, :] = (silu(x@swg) * (x@swu)) @ swd
// x-tile staged into LDS by the Tensor Data Mover (2D tile descriptor).
// ---------------------------------------------------------------------------
__global__ __launch_bounds__(256)
void shared_expert_kernel(const float* __restrict__ x,
                          const float* __restrict__ swg,
                          const float* __restrict__ swu,
                          const float* __restrict__ swd,
                          float* __restrict__ out) {
  __shared__ float xs[16][CHUNK + 4];
  __shared__ float act[16][II + 4];

  const int t0    = blockIdx.x * 16;
  const int tid   = threadIdx.x;
  const int wave  = tid >> 5;
  const int lane  = tid & 31;
  const int lhalf = lane >> 4;   // 0: K rows {0,1}; 1: K rows {2,3}
  const int lmod  = lane & 15;

#if USE_TDM
  const unsigned xs_lds = (unsigned)(size_t)(&xs[0][0]);
  // pad_interval code 6 => 128 DWORDs, pad_amount code 3 => 4 DWORDs
  const v8i g1 = tdm_group1_2d(HH, TT, CHUNK, 16, HH, 6u, 3u, true);
  const v4i gz = {0, 0, 0, 0};
#endif

  // ---- stage 1: act[16][II] = silu(x@swg) * (x@swu) ----
  for (int j = 0; j < 6; ++j) {            // 6 N-tiles of 16 per wave
    const int i0 = (wave * 6 + j) * 16;
    v8f g = V8F_ZERO, u = V8F_ZERO;
    for (int kc = 0; kc < HH; kc += CHUNK) {
      __syncthreads();                      // prior readers done with xs
#if USE_TDM
      if (wave == 0) {
        tdm_issue(xs_lds, x + (size_t)t0 * HH + kc, 0u, g1, gz, gz);
        __builtin_amdgcn_s_wait_tensorcnt(0);
      }
#else
      for (int l = tid; l < 16 * (CHUNK / 4); l += 256) {
        int row = l >> 5;
        int c4  = l & 31;
        *(float4*)&xs[row][c4 * 4] =
            *(const float4*)(x + (size_t)(t0 + row) * HH + kc + c4 * 4);
      }
#endif
      __syncthreads();                      // xs visible to all waves
      if (kc + CHUNK < HH) {                // prefetch next weight chunk rows
        __builtin_prefetch(&swg[(size_t)(kc + CHUNK + lane) * II + i0], 0, 0);
        __builtin_prefetch(&swu[(size_t)(kc + CHUNK + lane) * II + i0], 0, 0);
      }
      for (int k = 0; k < CHUNK; k += 4) {
        v2f a;
        a.x = xs[lmod][k + 2 * lhalf];
        a.y = xs[lmod][k + 2 * lhalf + 1];
        const size_t kb = (size_t)(kc + k + 2 * lhalf);
        v2f bg, bu;
        bg.x = swg[kb * II + i0 + lmod];
        bg.y = swg[(kb + 1) * II + i0 + lmod];
        bu.x = swu[kb * II + i0 + lmod];
        bu.y = swu[(kb + 1) * II + i0 + lmod];
        g = wmma4(a, bg, g);
        u = wmma4(a, bu, u);
      }
    }
    for (int r = 0; r < 8; ++r) {
      int   M  = r + lhalf * 8;
      float gv = g[r], uv = u[r];
      act[M][i0 + lmod] = (gv * sigmoid_f(gv)) * uv;
    }
  }
  __syncthreads();

  // ---- stage 2: out = act @ swd ----
  for (int jj = 0; jj < 16; ++jj) {
    const int n0 = (wave * 16 + jj) * 16;
    v8f acc = V8F_ZERO;
    for (int k = 0; k < II; k += 4) {
      v2f a;
      a.x = act[lmod][k + 2 * lhalf];
      a.y = act[lmod][k + 2 * lhalf + 1];
      const size_t kb = (size_t)(k + 2 * lhalf);
      v2f b;
      b.x = swd[kb * HH + n0 + lmod];
      b.y = swd[(kb + 1) * HH + n0 + lmod];
      acc = wmma4(a, b, acc);
    }
    for (int r = 0; r < 8; ++r) {
      int M = r + lhalf * 8;
      out[(size_t)(t0 + M) * HH + n0 + lmod] = acc[r];
    }
  }
}

// ---------------------------------------------------------------------------
// Routed experts: block (e, mblk) processes 16 gathered tokens of expert e.
// Gathered x rows staged by TDM *gather mode* (16-bit row indices in D#
// groups 2/3). Weighted down-projection atomicAdd'ed into out.
// ---------------------------------------------------------------------------
__global__ __launch_bounds__(256)
void routed_expert_kernel(const float* __restrict__ x,
                          const float* __restrict__ w_gate,
                          const float* __restrict__ w_up,
                          const float* __restrict__ w_down,
                          const int* __restrict__ expert_count,
                          const int* __restrict__ tok_list,
                          const float* __restrict__ wt_list,
                          float* __restrict__ out) {
  __shared__ float xs[16][CHUNK + 4];
  __shared__ float act[16][II + 4];
  __shared__ int   stok[16];
  __shared__ float swt[16];

  const int e    = blockIdx.x;
  const int base = blockIdx.y * 16;
  const int cnt  = expert_count[e];
  if (base >= cnt) return;                 // uniform across block

  const int tid   = threadIdx.x;
  const int wave  = tid >> 5;
  const int lane  = tid & 31;
  const int lhalf = lane >> 4;
  const int lmod  = lane & 15;

  if (tid < 16) {
    int r = base + tid;
    if (r < cnt) {
      stok[tid] = tok_list[(size_t)e * TT + r];
      swt[tid]  = wt_list[(size_t)e * TT + r];
    } else {
      stok[tid] = 0;                       // pad with token 0, weight 0
      swt[tid]  = 0.0f;
    }
  }
  __syncthreads();

  const float* wg = w_gate + (size_t)e * HH * II;
  const float* wu = w_up   + (size_t)e * HH * II;
  const float* wd = w_down + (size_t)e * II * HH;

#if USE_TDM
  const unsigned xs_lds = (unsigned)(size_t)(&xs[0][0]);
  const v8i g1 = tdm_group1_2d(HH, TT, CHUNK, 16, HH, 6u, 3u, true);
  v4i gi2, gi3;                            // 16 x 16-bit gather row indices
  for (int q = 0; q < 4; ++q) {
    gi2[q] = (stok[2 * q] & 0xFFFF) | (stok[2 * q + 1] << 16);
    gi3[q] = (stok[8 + 2 * q] & 0xFFFF) | (stok[8 + 2 * q + 1] << 16);
  }
#endif

  // ---- stage 1: act = silu(xg @ wg) * (xg @ wu), gathered token rows ----
  for (int j = 0; j < 6; ++j) {
    const int i0 = (wave * 6 + j) * 16;
    v8f g = V8F_ZERO, u = V8F_ZERO;
    for (int kc = 0; kc < HH; kc += CHUNK) {
      __syncthreads();
#if USE_TDM
      if (wave == 0) {
        tdm_issue(xs_lds, x + kc, /*gather_mode=*/0x80000000u, g1, gi2, gi3);
        __builtin_amdgcn_s_wait_tensorcnt(0);
      }
#else
      for (int l = tid; l < 16 * (CHUNK / 4); l += 256) {
        int row = l >> 5;
        int c4  = l & 31;
        *(float4*)&xs[row][c4 * 4] =
            *(const float4*)(x + (size_t)stok[row] * HH + kc + c4 * 4);
      }
#endif
      __syncthreads();
      if (kc + CHUNK < HH) {               // prefetch next weight chunk rows
        __builtin_prefetch(&wg[(size_t)(kc + CHUNK + lane) * II + i0], 0, 0);
        __builtin_prefetch(&wu[(size_t)(kc + CHUNK + lane) * II + i0], 0, 0);
      }
      for (int k = 0; k < CHUNK; k += 4) {
        v2f a;
        a.x = xs[lmod][k + 2 * lhalf];
        a.y = xs[lmod][k + 2 * lhalf + 1];
        const size_t kb = (size_t)(kc + k + 2 * lhalf);
        v2f bg, bu;
        bg.x = wg[kb * II + i0 + lmod];
        bg.y = wg[(kb + 1) * II + i0 + lmod];
        bu.x = wu[kb * II + i0 + lmod];
        bu.y = wu[(kb + 1) * II + i0 + lmod];
        g = wmma4(a, bg, g);
        u = wmma4(a, bu, u);
      }
    }
    for (int r = 0; r < 8; ++r) {
      int   M  = r + lhalf * 8;
      float gv = g[r], uv = u[r];
      act[M][i0 + lmod] = (gv * sigmoid_f(gv)) * uv;
    }
  }
  __syncthreads();

  // ---- stage 2: out[tok] += wt * (act @ wd) ----
  for (int jj = 0; jj < 16; ++jj) {
    const int n0 = (wave * 16 + jj) * 16;
    v8f acc = V8F_ZERO;
    for (int k = 0; k < II; k += 4) {
      v2f a;
      a.x = act[lmod][k + 2 * lhalf];
      a.y = act[lmod][k + 2 * lhalf + 1];
      const size_t kb = (size_t)(k + 2 * lhalf);
      v2f b;
      b.x = wd[kb * HH + n0 + lmod];
      b.y = wd[(kb + 1) * HH + n0 + lmod];
      acc = wmma4(a, b, acc);
    }
    for (int r = 0; r < 8; ++r) {
      int   M = r + lhalf * 8;
      float w = swt[M];
      if (w != 0.0f) {
        atomicAdd(&out[(size_t)stok[M] * HH + n0 + lmod], acc[r] * w);
      }
    }
  }
}

// ---------------------------------------------------------------------------
// Launch
// ---------------------------------------------------------------------------
extern "C" void kernel_launch(void* const* d_in, const int* in_sizes, int n_in,
                              void* d_out, int out_size, void* d_ws, size_t ws_size,
                              hipStream_t stream) {
  const float* x       = (const float*)d_in[0];
  const float* gate_w  = (const float*)d_in[1];
  const float* gate_b  = (const float*)d_in[2];
  const float* w_gate  = (const float*)d_in[3];
  const float* w_up    = (const float*)d_in[4];
  const float* w_down  = (const float*)d_in[5];
  const float* sw_gate = (const float*)d_in[6];
  const float* sw_up   = (const float*)d_in[7];
  const float* sw_down = (const float*)d_in[8];
  float* out = (float*)d_out;

  // Workspace layout
  char*  ws           = (char*)d_ws;
  float* scores       = (float*)(ws);                              // T*E f32
  int*   expert_count = (int*)  (ws + (size_t)TT * EE * 4);        // E i32
  int*   tok_list     = (int*)  (ws + (size_t)TT * EE * 4 + 1024); // E*T i32
  float* wt_list      = (float*)(ws + (size_t)TT * EE * 4 + 1024 +
                                 (size_t)EE * TT * 4);             // E*T f32

  zero_counts_kernel<<<1, 64, 0, stream>>>(expert_count);
  router_scores_kernel<<<TT, EE, 0, stream>>>(x, gate_w, scores);
  router_topk_kernel<<<(TT + 255) / 256, 256, 0, stream>>>(
      scores, gate_b, expert_count, tok_list, wt_list);
  // Shared expert initializes out (plain stores), routed experts accumulate.
  shared_expert_kernel<<<TT / 16, 256, 0, stream>>>(
      x, sw_gate, sw_up, sw_down, out);
  routed_expert_kernel<<<dim3(EE, TT / 16), 256, 0, stream>>>(
      x, w_gate, w_up, w_down, expert_count, tok_list, wt_list, out);
}